// GCN_23390391894412
// MI455X (gfx1250) — compile-verified
//
#include <hip/hip_runtime.h>

typedef float v8f __attribute__((ext_vector_type(8)));
typedef float v2f __attribute__((ext_vector_type(2)));

// ---------------------------------------------------------------------------
// Utility kernels
// ---------------------------------------------------------------------------
__global__ void gcn_fill_f32(float* __restrict__ p, float v, int n) {
    int i = blockIdx.x * blockDim.x + threadIdx.x;
    int stride = gridDim.x * blockDim.x;
    for (; i < n; i += stride) p[i] = v;
}

__global__ void gcn_degree_accum(const int* __restrict__ col, float* __restrict__ deg, int E) {
    int i = blockIdx.x * blockDim.x + threadIdx.x;
    int stride = gridDim.x * blockDim.x;
    for (; i < E; i += stride) atomicAdd(&deg[col[i]], 1.0f);
}

__global__ void gcn_rsqrt_inplace(float* __restrict__ d, int n) {
    int i = blockIdx.x * blockDim.x + threadIdx.x;
    int stride = gridDim.x * blockDim.x;
    for (; i < n; i += stride) {
        float v = d[i];
        d[i] = (v > 0.0f) ? rsqrtf(v) : 0.0f;
    }
}

// ---------------------------------------------------------------------------
// Dense transform: H[M x N] = X[M x K] * W[K x N], fp32 WMMA 16x16x4.
// One wave computes a 16 x N stripe; NT = N/16 accumulator tiles.
// M must be a multiple of 16 (50000 = 16*3125), N in {128, 64}, K = 128.
// A-fragment layout (ISA 7.12.2, 32-bit A 16x4): lanes 0-15 hold K=0 (v0),
// K=1 (v1); lanes 16-31 hold K=2 (v0), K=3 (v1). B mirrored over N-lanes.
// D: VGPR r, lane<16 -> row M=r; lane>=16 -> row M=8+r.
// ---------------------------------------------------------------------------
template <int NT>
__launch_bounds__(256)
__global__ void gcn_gemm_wmma(const float* __restrict__ X,
                              const float* __restrict__ W,
                              float* __restrict__ H,
                              int M, int K) {
    constexpr int N = NT * 16;
    const int wave = blockIdx.x * (blockDim.x >> 5) + (threadIdx.x >> 5);
    const int m0 = wave * 16;
    if (m0 >= M) return;                 // wave-uniform: EXEC all-ones for WMMA

    const int lane = threadIdx.x & 31;
    const int half = lane >> 4;          // 0 or 1
    const int lid  = lane & 15;

    v8f acc[NT] = {};

    const float* xrow = X + (size_t)(m0 + lid) * K;
    for (int k0 = 0; k0 < K; k0 += 4) {
        // A fragment: row (m0+lid), columns k0 + 2*half, k0 + 2*half + 1
        const float* ap = xrow + k0 + 2 * half;
        v2f a;
        a.x = ap[0];
        a.y = ap[1];
        const float* bbase = W + (size_t)(k0 + 2 * half) * N + lid;
#pragma unroll
        for (int t = 0; t < NT; ++t) {
            v2f b;
            b.x = bbase[t * 16];
            b.y = bbase[t * 16 + N];
            acc[t] = __builtin_amdgcn_wmma_f32_16x16x4_f32(
                false, a, false, b, (short)0, acc[t], false, false);
        }
    }

    // Store: lane covers column t*16+lid, VGPR r covers row m0 + half*8 + r
    float* orow = H + (size_t)(m0 + half * 8) * N + lid;
#pragma unroll
    for (int t = 0; t < NT; ++t) {
        float* op = orow + t * 16;
#pragma unroll
        for (int r = 0; r < 8; ++r) op[(size_t)r * N] = acc[t][r];
    }
}

// ---------------------------------------------------------------------------
// Edge scatter: out[col[e]] += H[row[e]] * dinv[row[e]] * dinv[col[e]]
// One lane handles 4 consecutive features (float4 gather + 4 f32 atomics).
// F = 128 -> 32 lanes/edge; F = 64 -> 16 lanes/edge.
// ---------------------------------------------------------------------------
template <int F>
__launch_bounds__(256)
__global__ void gcn_scatter(const float* __restrict__ H,
                            const int* __restrict__ row,
                            const int* __restrict__ col,
                            const float* __restrict__ dinv,
                            float* __restrict__ out,
                            int E) {
    constexpr int LPE = F / 4;           // lanes per edge
    const int gtid = blockIdx.x * blockDim.x + threadIdx.x;
    const int e = gtid / LPE;
    const int f = (gtid % LPE) * 4;
    if (e >= E) return;

    const int r = row[e];
    const int c = col[e];
    const float nrm = dinv[r] * dinv[c];

    const float4 hv = *(const float4*)(H + (size_t)r * F + f);
    float* op = out + (size_t)c * F + f;
    atomicAdd(op + 0, hv.x * nrm);
    atomicAdd(op + 1, hv.y * nrm);
    atomicAdd(op + 2, hv.z * nrm);
    atomicAdd(op + 3, hv.w * nrm);
}

// ---------------------------------------------------------------------------
// Self-loop contribution + bias (+ optional ReLU), elementwise over N*F:
//   agg[i] = act(agg[i] + H[i] * dinv[node]^2 + b[f])
// ---------------------------------------------------------------------------
template <int F, bool RELU>
__global__ void gcn_selfloop_bias(const float* __restrict__ H,
                                  const float* __restrict__ dinv,
                                  const float* __restrict__ b,
                                  float* __restrict__ agg,
                                  int total) {
    int i = blockIdx.x * blockDim.x + threadIdx.x;
    int stride = gridDim.x * blockDim.x;
    for (; i < total; i += stride) {
        const int node = i / F;
        const int f = i - node * F;
        const float d = dinv[node];
        float v = agg[i] + H[i] * d * d + b[f];
        if (RELU) v = fmaxf(v, 0.0f);
        agg[i] = v;
    }
}

// ---------------------------------------------------------------------------
// Host-side orchestration
// ---------------------------------------------------------------------------
extern "C" void kernel_launch(void* const* d_in, const int* in_sizes, int n_in,
                              void* d_out, int out_size, void* d_ws, size_t ws_size,
                              hipStream_t stream) {
    const float* x  = (const float*)d_in[0];
    const int*   ei = (const int*)d_in[1];     // [2, E] int32 (JAX x64 off)
    const float* W1 = (const float*)d_in[2];
    const float* b1 = (const float*)d_in[3];
    const float* W2 = (const float*)d_in[4];
    const float* b2 = (const float*)d_in[5];

    const int N = in_sizes[0] / 128;           // 50000 nodes
    const int E = in_sizes[1] / 2;             // 500000 edges
    const int* row = ei;
    const int* col = ei + E;

    // Workspace layout (256B aligned)
    char* base = (char*)d_ws;
    size_t off = 0;
    auto take = [&](size_t bytes) {
        char* p = base + off;
        off += (bytes + 255) & ~(size_t)255;
        return p;
    };
    float* deg  = (float*)take((size_t)N * 4);          // -> dinv in place
    float* h1   = (float*)take((size_t)N * 128 * 4);    // x @ W1
    float* agg1 = (float*)take((size_t)N * 128 * 4);    // layer-1 aggregate / relu out
    float* h2   = (float*)take((size_t)N * 64 * 4);     // relu_out @ W2
    float* out  = (float*)d_out;                        // layer-2 aggregate + final

    const int TPB = 256;
    auto blocks = [&](long long work) { return (int)((work + TPB - 1) / TPB); };

    // 1. Degree (init to 1.0 for self-loops), accumulate edges, rsqrt in place
    gcn_fill_f32<<<blocks(N), TPB, 0, stream>>>(deg, 1.0f, N);
    gcn_degree_accum<<<blocks(E), TPB, 0, stream>>>(col, deg, E);
    gcn_rsqrt_inplace<<<blocks(N), TPB, 0, stream>>>(deg, N);  // deg -> dinv
    const float* dinv = deg;

    // 2. Layer 1: h1 = x @ W1 (WMMA), zero agg, scatter, self-loop+bias+ReLU
    {
        const int waves = N / 16;                        // 3125
        const int grid = (waves + 7) / 8;                // 8 waves / 256-thread block
        gcn_gemm_wmma<8><<<grid, TPB, 0, stream>>>(x, W1, h1, N, 128);
    }
    gcn_fill_f32<<<blocks((long long)N * 128), TPB, 0, stream>>>(agg1, 0.0f, N * 128);
    gcn_scatter<128><<<blocks((long long)E * 32), TPB, 0, stream>>>(h1, row, col, dinv, agg1, E);
    gcn_selfloop_bias<128, true><<<blocks((long long)N * 128), TPB, 0, stream>>>(
        h1, dinv, b1, agg1, N * 128);

    // 3. Layer 2: h2 = agg1 @ W2 (WMMA), zero d_out, scatter into d_out, finish
    {
        const int waves = N / 16;
        const int grid = (waves + 7) / 8;
        gcn_gemm_wmma<4><<<grid, TPB, 0, stream>>>(agg1, W2, h2, N, 128);
    }
    gcn_fill_f32<<<blocks((long long)N * 64), TPB, 0, stream>>>(out, 0.0f, N * 64);
    gcn_scatter<64><<<blocks((long long)E * 16), TPB, 0, stream>>>(h2, row, col, dinv, out, E);
    gcn_selfloop_bias<64, false><<<blocks((long long)N * 64), TPB, 0, stream>>>(
        h2, dinv, b2, out, N * 64);

    (void)n_in; (void)out_size; (void)ws_size;
}